// SSD_72378788873060
// MI455X (gfx1250) — compile-verified
//
#include <hip/hip_runtime.h>
#include <hip/hip_bf16.h>

#define BATCH   16
#define NPRIOR  8732
#define NCLS    81
#define TOPK    200
#define CONF_T  0.01f
#define NMS_T   0.45f
#define VAR0    0.1f
#define VAR1    0.2f

#define GROUPS_PER_B ((NPRIOR + 15) / 16)      /* 546 groups of 16 priors */
#define TOTAL_GROUPS (BATCH * GROUPS_PER_B)    /* 8736 waves total        */
#define SM_BLOCKS    (TOTAL_GROUPS / 8)        /* 8 waves per 256-block   */

typedef float v2f __attribute__((ext_vector_type(2)));
typedef float v8f __attribute__((ext_vector_type(8)));
typedef int   vec4i __attribute__((vector_size(16)));   // matches builtin param type
typedef vec4i __attribute__((address_space(1))) *gptr4; // global int4*
typedef vec4i __attribute__((address_space(3))) *lptr4; // LDS int4*

#if defined(__has_builtin)
#if __has_builtin(__builtin_amdgcn_global_load_async_to_lds_b128) && \
    __has_builtin(__builtin_amdgcn_s_wait_asynccnt)
#define HAVE_ASYNC_LDS 1
#endif
#endif

// ---------------------------------------------------------------------------
// Kernel 1: SSD box decode  loc[B,N,4] + anchors[N,4] -> decoded[B,N,4] xyxy
// ---------------------------------------------------------------------------
__global__ __launch_bounds__(256) void ssd_decode(
    const float* __restrict__ loc, const float* __restrict__ anchors,
    float* __restrict__ decoded)
{
  int tid = blockIdx.x * blockDim.x + threadIdx.x;
  if (tid >= BATCH * NPRIOR) return;
  int n = tid % NPRIOR;
  const float* l = loc + (size_t)tid * 4;
  const float* a = anchors + (size_t)n * 4;
  float cx = a[0] + l[0] * VAR0 * a[2];
  float cy = a[1] + l[1] * VAR0 * a[3];
  float w  = a[2] * __expf(l[2] * VAR1);
  float h  = a[3] * __expf(l[3] * VAR1);
  float* o = decoded + (size_t)tid * 4;
  o[0] = cx - 0.5f * w;
  o[1] = cy - 0.5f * h;
  o[2] = cx + 0.5f * w;
  o[3] = cy + 0.5f * h;
}

// ---------------------------------------------------------------------------
// Kernel 2: softmax over C=81 with WMMA row-sum; writes transposed [B,C,N].
// One wave = 16 priors. Lane r holds classes {4t,4t+1}, lane r+16 holds
// {4t+2,4t+3} of row r -- exactly the 16x4 f32 A-matrix layout. Row sums are
// computed as D = A x ones via V_WMMA_F32_16X16X4_F32 accumulated over 21
// K-chunks (81 classes, zero padded).
// ---------------------------------------------------------------------------
__global__ __launch_bounds__(256) void ssd_softmax_wmma(
    const float* __restrict__ conf, float* __restrict__ probsT)
{
  const int lane = threadIdx.x & 31;
  const int wgrp = blockIdx.x * 8 + (threadIdx.x >> 5);
  const int b    = wgrp / GROUPS_PER_B;
  const int g    = wgrp % GROUPS_PER_B;
  const int r    = lane & 15;
  const int n    = g * 16 + r;
  const bool valid = (n < NPRIOR);
  const int  coff  = (lane < 16) ? 0 : 2;
  const size_t base = ((size_t)b * NPRIOR + (size_t)n) * NCLS;

  const float NEG_INF = -__builtin_inff();
  float x0[21], x1[21];
  float m = NEG_INF;
#pragma unroll
  for (int t = 0; t < 21; ++t) {
    int c0 = 4 * t + coff;
    int c1 = c0 + 1;
    bool ok0 = valid && (c0 < NCLS);
    bool ok1 = valid && (c1 < NCLS);
    float a0 = ok0 ? conf[base + c0] : NEG_INF;
    float a1 = ok1 ? conf[base + c1] : NEG_INF;
    x0[t] = a0; x1[t] = a1;
    m = fmaxf(m, fmaxf(a0, a1));
  }
  // the two lanes holding row r are lane r and lane r+16
  m = fmaxf(m, __shfl_xor(m, 16, 32));

  v8f acc = {0.f, 0.f, 0.f, 0.f, 0.f, 0.f, 0.f, 0.f};
  v2f bones; bones[0] = 1.0f; bones[1] = 1.0f;   // B = ones(4x16)
#pragma unroll
  for (int t = 0; t < 21; ++t) {
    int c0 = 4 * t + coff;
    bool ok0 = valid && (c0 < NCLS);
    bool ok1 = valid && (c0 + 1 < NCLS);
    float e0 = ok0 ? __expf(x0[t] - m) : 0.0f;
    float e1 = ok1 ? __expf(x1[t] - m) : 0.0f;
    x0[t] = e0; x1[t] = e1;
    v2f a; a[0] = e0; a[1] = e1;
    // D[r][*] = rowsum(E_chunk) accumulated;  emits v_wmma_f32_16x16x4_f32
    acc = __builtin_amdgcn_wmma_f32_16x16x4_f32(
        false, a, false, bones, (short)0, acc, false, false);
  }

  // D layout: rows 0..7 in VGPR0..7 of lanes 0..15, rows 8..15 in lanes 16..31.
  // Each lane wants rowsum(lane&15): 8-way select then cross-half shuffle.
  int i3 = lane & 7;
  float s = acc[0];
  s = (i3 == 1) ? acc[1] : s;
  s = (i3 == 2) ? acc[2] : s;
  s = (i3 == 3) ? acc[3] : s;
  s = (i3 == 4) ? acc[4] : s;
  s = (i3 == 5) ? acc[5] : s;
  s = (i3 == 6) ? acc[6] : s;
  s = (i3 == 7) ? acc[7] : s;
  int q = lane >> 3;                         // 0..3
  int src = lane + ((q == 1) ? 8 : (q == 2) ? -16 : 0);
  s = __shfl(s, src, 32);
  float inv = 1.0f / s;

#pragma unroll
  for (int t = 0; t < 21; ++t) {
    int c0 = 4 * t + coff;
    int c1 = c0 + 1;
    if (valid && c0 < NCLS)
      probsT[((size_t)(b * NCLS + c0)) * NPRIOR + n] = x0[t] * inv;
    if (valid && c1 < NCLS)
      probsT[((size_t)(b * NCLS + c1)) * NPRIOR + n] = x1[t] * inv;
  }
}

// ---------------------------------------------------------------------------
// Kernel 3: one block per (b,c). Scores staged global->LDS with
// GLOBAL_LOAD_ASYNC_TO_LDS_B128 (ASYNCcnt + s_wait_asynccnt) when the
// toolchain exposes the builtins; sync fallback otherwise. Then exact top-200
// via 64-bit key bit-bisection over LDS, bitonic sort, sequential NMS,
// masked output write. Class 0 -> zeros.
// key = (score_bits << 32) | ~index : distinct keys, desc score, asc index.
// ---------------------------------------------------------------------------
__global__ __launch_bounds__(256) void ssd_topk_nms(
    const float* __restrict__ probsT, const float* __restrict__ decoded,
    float* __restrict__ out)
{
  __shared__ __align__(16) unsigned scoreU[NPRIOR];
  __shared__ unsigned long long cand[256];
  __shared__ float bx[TOPK][4];
  __shared__ float sc[TOPK];
  __shared__ int   kp[TOPK];
  __shared__ int   wsum[8];
  __shared__ int   cnt;

  const int t = threadIdx.x;
  const int b = blockIdx.x / NCLS;
  const int c = blockIdx.x % NCLS;

  if (c == 0) {  // background class: zero rows (uniform branch per block)
    size_t o = ((size_t)b * NCLS) * TOPK * 5;
    for (int i = t; i < TOPK * 5; i += 256) out[o + i] = 0.0f;
    return;
  }

  const float* col = probsT + ((size_t)(b * NCLS + c)) * NPRIOR;

#if HAVE_ASYNC_LDS
  // DMA the 8732-float column straight into LDS, 16B chunks (8732 = 4*2183).
  for (int j = t; j < NPRIOR / 4; j += 256) {
    __builtin_amdgcn_global_load_async_to_lds_b128(
        (gptr4)(col + 4 * j), (lptr4)(&scoreU[4 * j]), 0, 0);
  }
  __builtin_amdgcn_s_wait_asynccnt(0);
  __syncthreads();
  // in-place confidence mask (raw bytes landed in LDS)
  for (int i = t; i < NPRIOR; i += 256) {
    float p = __uint_as_float(scoreU[i]);
    scoreU[i] = __float_as_uint(p > CONF_T ? p : 0.0f);
  }
  __syncthreads();
#else
  for (int i = t; i < NPRIOR; i += 256) {
    __builtin_prefetch(col + i + 4096, 0, 1);   // global_prefetch_b8
    float p = col[i];
    p = (p > CONF_T) ? p : 0.0f;                // masked scores (>=0)
    scoreU[i] = __float_as_uint(p);
  }
  __syncthreads();
#endif

  // bit-bisection: largest tau with count(key >= tau) >= TOPK == 200th key
  unsigned long long tau = 0ull;
  for (int bit = 63; bit >= 0; --bit) {
    unsigned long long trial = tau | (1ull << bit);
    int local = 0;
    for (int i = t; i < NPRIOR; i += 256) {
      unsigned long long key =
          ((unsigned long long)scoreU[i] << 32) | (unsigned)(~i);
      local += (key >= trial) ? 1 : 0;
    }
#pragma unroll
    for (int o = 16; o > 0; o >>= 1) local += __shfl_xor(local, o, 32);
    if ((t & 31) == 0) wsum[t >> 5] = local;
    __syncthreads();
    int total = 0;
#pragma unroll
    for (int w = 0; w < 8; ++w) total += wsum[w];
    if (total >= TOPK) tau = trial;
    __syncthreads();
  }

  // gather the exactly-200 selected keys
  if (t == 0) cnt = 0;
  cand[t] = 0ull;
  __syncthreads();
  for (int i = t; i < NPRIOR; i += 256) {
    unsigned long long key =
        ((unsigned long long)scoreU[i] << 32) | (unsigned)(~i);
    if (key >= tau) {
      int p = atomicAdd(&cnt, 1);
      if (p < 256) cand[p] = key;
    }
  }
  __syncthreads();

  // bitonic sort 256 keys, descending (zero padding sinks to the tail)
  for (int k = 2; k <= 256; k <<= 1) {
    for (int j = k >> 1; j > 0; j >>= 1) {
      int ixj = t ^ j;
      if (ixj > t) {
        unsigned long long a = cand[t], d = cand[ixj];
        bool up = ((t & k) == 0);               // "up" segment = descending
        if (up ? (a < d) : (a > d)) { cand[t] = d; cand[ixj] = a; }
      }
      __syncthreads();
    }
  }

  // unpack top-200, fetch decoded boxes, init keep mask
  if (t < TOPK) {
    unsigned long long key = cand[t];
    float scv = __uint_as_float((unsigned)(key >> 32));
    unsigned idx = ~(unsigned)key;
    if (idx >= NPRIOR) { idx = 0; scv = 0.0f; }   // defensive
    const float* bp = decoded + ((size_t)b * NPRIOR + idx) * 4;
    bx[t][0] = bp[0]; bx[t][1] = bp[1]; bx[t][2] = bp[2]; bx[t][3] = bp[3];
    sc[t] = scv;
    kp[t] = (scv > CONF_T) ? 1 : 0;
  }
  __syncthreads();

  // sequential NMS: iteration i suppresses j>i with IoU>thresh if keep[i]
  for (int i = 0; i < TOPK - 1; ++i) {
    if (t > i && t < TOPK && kp[i]) {
      float lx = fmaxf(bx[i][0], bx[t][0]);
      float ly = fmaxf(bx[i][1], bx[t][1]);
      float rx = fminf(bx[i][2], bx[t][2]);
      float ry = fminf(bx[i][3], bx[t][3]);
      float inter = fmaxf(rx - lx, 0.0f) * fmaxf(ry - ly, 0.0f);
      float ai = (bx[i][2] - bx[i][0]) * (bx[i][3] - bx[i][1]);
      float at = (bx[t][2] - bx[t][0]) * (bx[t][3] - bx[t][1]);
      float uni = fmaxf(ai + at - inter, 1e-12f);
      if (inter / uni > NMS_T) kp[t] = 0;
    }
    __syncthreads();
  }

  // masked write: (score, x1,y1,x2,y2) * keep
  if (t < TOPK) {
    float kf = kp[t] ? 1.0f : 0.0f;
    size_t o = (((size_t)b * NCLS + c) * TOPK + (size_t)t) * 5;
    out[o + 0] = sc[t] * kf;
    out[o + 1] = bx[t][0] * kf;
    out[o + 2] = bx[t][1] * kf;
    out[o + 3] = bx[t][2] * kf;
    out[o + 4] = bx[t][3] * kf;
  }
}

// ---------------------------------------------------------------------------
// Launch: d_in[0]=loc_heads[B,N,4], d_in[1]=conf_heads[B,N,C], d_in[2]=anchors
// ws: decoded (B*N*4 floats) then probsT (B*C*N floats) -> ~47.5 MB total.
// ---------------------------------------------------------------------------
extern "C" void kernel_launch(void* const* d_in, const int* in_sizes, int n_in,
                              void* d_out, int out_size, void* d_ws, size_t ws_size,
                              hipStream_t stream) {
  const float* loc     = (const float*)d_in[0];
  const float* conf    = (const float*)d_in[1];
  const float* anchors = (const float*)d_in[2];
  float* outp = (float*)d_out;

  float* decoded = (float*)d_ws;
  float* probsT  = decoded + (size_t)BATCH * NPRIOR * 4;

  int decN = BATCH * NPRIOR;
  ssd_decode<<<(decN + 255) / 256, 256, 0, stream>>>(loc, anchors, decoded);
  ssd_softmax_wmma<<<SM_BLOCKS, 256, 0, stream>>>(conf, probsT);
  ssd_topk_nms<<<BATCH * NCLS, 256, 0, stream>>>(probsT, decoded, outp);
}